// QINCoStep_85306640433320
// MI455X (gfx1250) — compile-verified
//
#include <hip/hip_runtime.h>

typedef __attribute__((ext_vector_type(2))) float v2f;
typedef __attribute__((ext_vector_type(8))) float v8f;

#define BN 32768
#define KC 4096
#define DD 256

#define MBLK 64      // rows per workgroup
#define NBLK 128     // codewords per N-iteration
#define DCHUNK 64    // D columns per B-staging chunk
#define PA 260       // A LDS pitch (floats), pad 4 -> bank rotate, 16B aligned rows
#define PB 68        // B LDS pitch (floats), pad 4 -> bank rotate, 16B aligned rows
#define NCHUNK ((KC / NBLK) * (DD / DCHUNK))   // 128 total pipeline steps

// Async global->LDS copy, 16B per lane, tracked with ASYNCcnt (CDNA5).
__device__ __forceinline__ void async_copy_b128(unsigned lds_addr, const void* gaddr) {
  asm volatile("global_load_async_to_lds_b128 %0, %1, off"
               :: "v"(lds_addr), "v"(gaddr)
               : "memory");
}
__device__ __forceinline__ void wait_asynccnt0() {
  asm volatile("s_wait_asynccnt 0" ::: "memory");
}
__device__ __forceinline__ unsigned lds_off(const void* p) {
  // Generic LDS pointer: high 32 bits are the shared aperture, low 32 bits the
  // LDS byte offset (aperture check uses ADDR[63:32] only).
  return (unsigned)(unsigned long long)p;
}

// ---------------- kernel 0: c_sq[k] = ||codebook[k]||^2 ----------------
__global__ __launch_bounds__(256) void csq_kernel(const float* __restrict__ cb,
                                                  float* __restrict__ csq) {
  const int wid = threadIdx.x >> 5;
  const int lane = threadIdx.x & 31;
  const int row = blockIdx.x * 8 + wid;            // 4096 rows, 8 waves/block
  const float4* p = (const float4*)(cb + (size_t)row * DD);
  float4 a = p[lane * 2];
  float4 b = p[lane * 2 + 1];
  float s = a.x * a.x + a.y * a.y + a.z * a.z + a.w * a.w +
            b.x * b.x + b.y * b.y + b.z * b.z + b.w * b.w;
#pragma unroll
  for (int off = 16; off; off >>= 1) s += __shfl_xor(s, off, 32);
  if (lane == 0) csq[row] = s;
}

// ---------------- kernel 1: fused WMMA cross-term + argmin ----------------
__global__ __launch_bounds__(256) void argmin_kernel(const float* __restrict__ x,
                                                     const float* __restrict__ cb,
                                                     const float* __restrict__ csq,
                                                     int* __restrict__ codes) {
  __shared__ __align__(16) float As[MBLK * PA];
  __shared__ __align__(16) float Bs[2][NBLK * PB];
  __shared__ float redMin[MBLK][4];
  __shared__ int   redIdx[MBLK][4];

  const int t   = threadIdx.x;
  const int wid = t >> 5;
  const int wm  = wid >> 2;         // 0..1  (M wave)
  const int wn  = wid & 3;          // 0..3  (N wave)
  const int l   = t & 31;
  const int h   = l >> 4;           // lane half
  const int lm  = l & 15;
  const int blockM = blockIdx.x * MBLK;

  // Stage A block (64 x 256 fp32, full D) into LDS once (plain path; one-time).
#pragma unroll
  for (int j = 0; j < 16; ++j) {
    int idx = j * 256 + t;          // float4 index, 4096 total
    int row = idx >> 6;             // 64 float4 per row
    int c4  = idx & 63;
    float4 v = ((const float4*)(x + (size_t)(blockM + row) * DD))[c4];
    *(float4*)&As[row * PA + c4 * 4] = v;
  }

  // Per-thread B staging slice: 2048 float4 per chunk / 256 threads = 8 each.
  // Issue async DMA for chunk `ch` into buffer `buf`.
  auto issue_chunk = [&](int ch, int buf) {
    const int nt = (ch >> 2) * NBLK;
    const int d0 = (ch & 3) * DCHUNK;
#pragma unroll
    for (int j = 0; j < 8; ++j) {
      int idx = j * 256 + t;        // float4 index, 2048 total
      int row = idx >> 4;           // 16 float4 per row
      int c4  = idx & 15;
      const void* g = cb + (size_t)(nt + row) * DD + d0 + c4 * 4;
      async_copy_b128(lds_off(&Bs[buf][row * PB + c4 * 4]), g);
    }
  };

  v8f acc[2][2];
  float best[2][8];
  int   bidx[2][8];
#pragma unroll
  for (int mi = 0; mi < 2; ++mi)
#pragma unroll
    for (int i = 0; i < 8; ++i) { best[mi][i] = 3.402823e38f; bidx[mi][i] = 0; }
#pragma unroll
  for (int mi = 0; mi < 2; ++mi)
#pragma unroll
    for (int ni = 0; ni < 2; ++ni)
#pragma unroll
      for (int q = 0; q < 8; ++q) acc[mi][ni][q] = 0.0f;

  issue_chunk(0, 0);                // pipeline prologue

  for (int ch = 0; ch < NCHUNK; ++ch) {
    const int buf = ch & 1;
    const int d0  = (ch & 3) * DCHUNK;

    wait_asynccnt0();               // my chunk-ch DMAs done
    __syncthreads();                // everyone's chunk-ch in LDS; buf^1 free

    if (ch + 1 < NCHUNK) issue_chunk(ch + 1, buf ^ 1);  // overlap with compute

    const float* __restrict__ B = &Bs[buf][0];
#pragma unroll
    for (int kk = 0; kk < DCHUNK; kk += 4) {
      v2f a0 = *(const v2f*)&As[(wm * 32 +  0 + lm) * PA + d0 + kk + h * 2];
      v2f a1 = *(const v2f*)&As[(wm * 32 + 16 + lm) * PA + d0 + kk + h * 2];
      v2f b0 = *(const v2f*)&B[(wn * 32 +  0 + lm) * PB + kk + h * 2];
      v2f b1 = *(const v2f*)&B[(wn * 32 + 16 + lm) * PB + kk + h * 2];
      acc[0][0] = __builtin_amdgcn_wmma_f32_16x16x4_f32(
          false, a0, false, b0, (short)0, acc[0][0], false, false);
      acc[0][1] = __builtin_amdgcn_wmma_f32_16x16x4_f32(
          false, a0, false, b1, (short)0, acc[0][1], false, false);
      acc[1][0] = __builtin_amdgcn_wmma_f32_16x16x4_f32(
          false, a1, false, b0, (short)0, acc[1][0], false, false);
      acc[1][1] = __builtin_amdgcn_wmma_f32_16x16x4_f32(
          false, a1, false, b1, (short)0, acc[1][1], false, false);
    }

    if ((ch & 3) == 3) {
      // N-tile finished: s = c_sq[n] - 2*cross ; update running argmin.
      const int nt = (ch >> 2) * NBLK;
#pragma unroll
      for (int ni = 0; ni < 2; ++ni) {
        int n = nt + wn * 32 + ni * 16 + lm;
        float cs = csq[n];
#pragma unroll
        for (int mi = 0; mi < 2; ++mi)
#pragma unroll
          for (int i = 0; i < 8; ++i) {
            float s = cs - 2.0f * acc[mi][ni][i];
            if (s < best[mi][i]) { best[mi][i] = s; bidx[mi][i] = n; }
          }
      }
#pragma unroll
      for (int mi = 0; mi < 2; ++mi)
#pragma unroll
        for (int ni = 0; ni < 2; ++ni)
#pragma unroll
          for (int q = 0; q < 8; ++q) acc[mi][ni][q] = 0.0f;
    }
  }

  // Intra-wave reduce across the 16 N-lanes (tie-break: smaller index).
#pragma unroll
  for (int mi = 0; mi < 2; ++mi)
#pragma unroll
    for (int i = 0; i < 8; ++i) {
      float b_ = best[mi][i];
      int   ix = bidx[mi][i];
#pragma unroll
      for (int off = 8; off; off >>= 1) {
        float ob = __shfl_xor(b_, off, 32);
        int   oi = __shfl_xor(ix, off, 32);
        if (ob < b_ || (ob == b_ && oi < ix)) { b_ = ob; ix = oi; }
      }
      if (lm == 0) {
        int rl = wm * 32 + mi * 16 + h * 8 + i;   // local row 0..63
        redMin[rl][wn] = b_;
        redIdx[rl][wn] = ix;
      }
    }
  __syncthreads();

  // Cross-wave (4 N-waves) reduce, one thread per row.
  if (t < MBLK) {
    float b_ = redMin[t][0];
    int   ix = redIdx[t][0];
#pragma unroll
    for (int w = 1; w < 4; ++w) {
      float ob = redMin[t][w];
      int   oi = redIdx[t][w];
      if (ob < b_ || (ob == b_ && oi < ix)) { b_ = ob; ix = oi; }
    }
    codes[blockM + t] = ix;
  }
}

// ---------------- kernel 2: gather quant rows ----------------
__global__ __launch_bounds__(256) void gather_kernel(const float* __restrict__ cb,
                                                     const int* __restrict__ codes,
                                                     float* __restrict__ quant) {
  const int wid = threadIdx.x >> 5;
  const int lane = threadIdx.x & 31;
  const int row = blockIdx.x * 8 + wid;            // 32768 rows, wave per row
  const int code = codes[row];
  const float4* src = (const float4*)(cb + (size_t)code * DD);
  float4* dst = (float4*)(quant + (size_t)row * DD);
  dst[lane * 2]     = src[lane * 2];
  dst[lane * 2 + 1] = src[lane * 2 + 1];
}

extern "C" void kernel_launch(void* const* d_in, const int* in_sizes, int n_in,
                              void* d_out, int out_size, void* d_ws, size_t ws_size,
                              hipStream_t stream) {
  const float* residual = (const float*)d_in[0];
  const float* codebook = (const float*)d_in[1];
  float* out  = (float*)d_out;
  int*   codes = (int*)out;          // first BN elements: int32 codes
  float* quant = out + BN;           // next BN*DD floats: quantized rows
  float* csq   = (float*)d_ws;       // 4096 floats scratch

  csq_kernel<<<KC / 8, 256, 0, stream>>>(codebook, csq);
  argmin_kernel<<<BN / MBLK, 256, 0, stream>>>(residual, codebook, csq, codes);
  gather_kernel<<<BN / 8, 256, 0, stream>>>(codebook, codes, quant);
}